// GtNet_59957743452685
// MI455X (gfx1250) — compile-verified
//
#include <hip/hip_runtime.h>

typedef __attribute__((ext_vector_type(2))) float        v2f;
typedef __attribute__((ext_vector_type(8))) float        v8f;
typedef __attribute__((ext_vector_type(4))) unsigned int v4u;
typedef __attribute__((ext_vector_type(8))) int          v8i;
typedef __attribute__((ext_vector_type(4))) int          v4i;

#define BATCH   16
#define H       256
#define W       256
#define HW      (H * W)
#define MR      3
#define K7      7
#define NC      49            // K7*K7 classes / taps
#define KPAD    52            // K dim padded to multiple of 4 (WMMA f32 K=4)
#define NPAD    64            // N dim padded to 4 WMMA 16-wide tiles
#define PRED_ELEMS (BATCH * 3 * HW)

// ---------------------------------------------------------------------------
// Kernel 1: per-pixel axis weights (ax[7], ay[7]) from gt_motion, plus the
// m_mask output (outer product, written in NCHW layout).
// ---------------------------------------------------------------------------
__global__ __launch_bounds__(256) void axis_mask_kernel(
    const float* __restrict__ gt, float* __restrict__ mask_out,
    float* __restrict__ axay)
{
    int pix = blockIdx.x * 256 + threadIdx.x;
    if (pix >= BATCH * HW) return;
    int b = pix / HW;
    int r = pix - b * HW;                 // y*W + x

    float mx = gt[((size_t)b * 2 + 0) * HW + r];
    float my = gt[((size_t)b * 2 + 1) * HW + r];

    float fx = floorf(mx), frx = mx - fx; int ix = (int)fx + MR;
    float fy = floorf(my), fry = my - fy; int iy = (int)fy + MR;

    float ax[K7], ay[K7];
#pragma unroll
    for (int i = 0; i < K7; ++i) {
        ax[i] = (i == ix ? 1.0f - frx : 0.0f) + (i == ix + 1 ? frx : 0.0f);
        ay[i] = (i == iy ? 1.0f - fry : 0.0f) + (i == iy + 1 ? fry : 0.0f);
    }

    // workspace: 16 floats / pixel (7 ax, 7 ay, 2 pad) -> aligned, compact
    float* wsp = axay + (size_t)pix * 16;
#pragma unroll
    for (int i = 0; i < K7; ++i) { wsp[i] = ax[i]; wsp[K7 + i] = ay[i]; }
    wsp[14] = 0.0f; wsp[15] = 0.0f;

    // m_mask output, layout [b][n][y][x]
    float* mb = mask_out + (size_t)b * NC * HW + r;
#pragma unroll
    for (int ny = 0; ny < K7; ++ny) {
        float a = ay[ny];
#pragma unroll
        for (int nx = 0; nx < K7; ++nx)
            mb[(size_t)(ny * K7 + nx) * HW] = a * ax[nx];
    }
}

// ---------------------------------------------------------------------------
// Kernel 2: per 16x8 pixel block:
//   phase 0 (TDM):  async tensor DMA of the 49x49 kernel into LDS
//   phase 1 (VALU): depthwise out_mask o[p][n] from staged ax/ay halo tile
//   phase 2 (WMMA): W[p][t] = o[p][:] x Kmat  (16x52 @ 52x64, f32 WMMA)
//   phase 3 (VALU): pred[c,p] = sum_t im[c, p+t] * W[p][t]
// ---------------------------------------------------------------------------
__global__ __launch_bounds__(256) void pred_wmma_kernel(
    const float* __restrict__ im, const float* __restrict__ mker,
    const float* __restrict__ axay, float* __restrict__ pred)
{
    __shared__ float s_axay[14 * 22 * 14];       // halo tile of (ax,ay), 17.2 KB
    __shared__ float s_kmat[KPAD][NPAD];         // zero-padded 49x49 kernel, 13.3 KB
    __shared__ float s_acc[8][16][NPAD];         // per-wave: o then W, 32 KB
                                                 // (front 9.6 KB doubles as TDM staging)

    const int tid  = threadIdx.x;
    const int wv   = tid >> 5;                   // wave = block row (0..7)
    const int lane = tid & 31;
    const int x0   = blockIdx.x * 16;
    const int y0   = blockIdx.y * 8;
    const int b    = blockIdx.z;

    // ---- phase 0: TDM async load of raw 49x49 kernel (2401 f32, contiguous)
    // into LDS staging (aliases s_acc, which is only written in phase 1).
    float* kraw = &s_acc[0][0][0];
    if (wv == 0) {
        unsigned long long ga = (unsigned long long)(uintptr_t)mker;
        unsigned int ldsa = (unsigned int)(uintptr_t)kraw;   // LDS offset = addr[31:0]
        // D# group0: count=1 | lds_addr | global_addr[56:0] | type=2 (bits 127:126)
        v4u g0 = { 1u, ldsa, (unsigned int)ga,
                   (unsigned int)((ga >> 32) & 0x1FFFFFFu) | 0x80000000u };
        // D# group1: data_size=2 (4B); tensor_dim0=2401 (dw1[31:16]|dw2[15:0]);
        // tensor_dim1=1 (dw2[31:16]); tile_dim0=2401 (dw3[31:16]);
        // tile_dim1/2=0 (unused); tensor_dim0_stride=2401 (dw5)
        v8i g1 = { 0x00020000, 0x09610000, 0x00010000, 0x09610000,
                   0, 2401, 0, 0 };
        v4i gz4 = { 0, 0, 0, 0 };                // groups 2/3: <=2-D tensor
        v8i gz8 = { 0, 0, 0, 0, 0, 0, 0, 0 };    // unused trailing group
        __builtin_amdgcn_tensor_load_to_lds(g0, g1, gz4, gz4, gz8, 0);
        __builtin_amdgcn_s_wait_tensorcnt(0);
    }
    __syncthreads();

    // ---- expand Kmat (rows = class n, cols = tap t), zero padded to 52x64
    for (int i = tid; i < KPAD * NPAD; i += 256) {
        int k = i >> 6, t = i & 63;
        s_kmat[k][t] = (k < NC && t < NC) ? kraw[k * NC + t] : 0.0f;
    }

    // ---- stage ax/ay halo tile: rows y0-3..y0+10, cols x0-3..x0+18
    for (int i = tid; i < 14 * 22; i += 256) {
        int ry = i / 22, rx = i - ry * 22;
        int gy = y0 + ry - MR, gx = x0 + rx - MR;
        float* dst = &s_axay[i * 14];
        if ((unsigned)gy < H && (unsigned)gx < W) {
            const float* src = &axay[((size_t)b * HW + (size_t)gy * W + gx) * 16];
            __builtin_prefetch(src + 64, 0, 0);
#pragma unroll
            for (int j = 0; j < 14; ++j) dst[j] = src[j];
        } else {
#pragma unroll
            for (int j = 0; j < 14; ++j) dst[j] = 0.0f;
        }
    }
    __syncthreads();

    // ---- phase 1: depthwise conv -> o[p][n] into this wave's s_acc slice.
    // zero A-matrix K-padding cols first (WMMA propagates NaN through 0*NaN)
    for (int i = lane; i < 16 * 16; i += 32) {
        int p = i >> 4, c = 48 + (i & 15);
        s_acc[wv][p][c] = 0.0f;
    }
    for (int it = 0; it < 25; ++it) {
        int task = it * 32 + lane;               // 16 pixels x 49 classes
        int px = task & 15;
        int n  = task >> 4;
        if (n < NC) {
            int ny = n / K7, nx = n - ny * K7;
            float acc = 0.0f;
#pragma unroll
            for (int dy = 0; dy < K7; ++dy) {
                const float* arow = &s_axay[((wv + dy) * 22 + px) * 14];
#pragma unroll
                for (int dx = 0; dx < K7; ++dx) {
                    float axv = arow[dx * 14 + nx];
                    float ayv = arow[dx * 14 + K7 + ny];
                    acc = fmaf(axv * ayv, s_kmat[n][dy * K7 + dx], acc);
                }
            }
            s_acc[wv][px][n] = acc;
        }
    }
    __syncthreads();

    // ---- phase 2: W = o x Kmat via V_WMMA_F32_16X16X4_F32.
    // A (16x4 f32): lanes 0-15 M, vgpr0/1 = K,K+1; lanes 16-31 = K+2,K+3.
    // B (4x16 f32): symmetric lane->N, half-wave picks K pair.
    const int half = lane >> 4;
    const int pix  = lane & 15;
    v8f cacc[4] = {};
    for (int k0 = 0; k0 < KPAD; k0 += 4) {
        int kk = k0 + 2 * half;
        v2f a;
        a[0] = s_acc[wv][pix][kk];
        a[1] = s_acc[wv][pix][kk + 1];
#pragma unroll
        for (int t = 0; t < 4; ++t) {
            v2f bm;
            bm[0] = s_kmat[kk][t * 16 + pix];
            bm[1] = s_kmat[kk + 1][t * 16 + pix];
            cacc[t] = __builtin_amdgcn_wmma_f32_16x16x4_f32(
                false, a, false, bm, (short)0, cacc[t], false, false);
        }
    }
    // spill D (W[p][t]) back over this wave's s_acc slice
#pragma unroll
    for (int t = 0; t < 4; ++t)
#pragma unroll
        for (int v = 0; v < 8; ++v)
            s_acc[wv][v + 8 * half][t * 16 + pix] = cacc[t][v];
    __syncthreads();

    // ---- phase 3: spatially-varying 49-tap conv of the 3 image channels
    for (int rep = 0; rep < 2; ++rep) {
        int task = lane + rep * 32;              // 16 pixels x 3 channels = 48
        if (task < 48) {
            int px = task & 15;
            int c  = task >> 4;
            int y = y0 + wv, x = x0 + px;
            const float* imc = im + ((size_t)b * 6 + 3 + c) * HW;   // last 3 ch
            float acc = 0.0f;
#pragma unroll
            for (int dy = 0; dy < K7; ++dy) {
                int iy = y + dy - MR;
                if ((unsigned)iy < H) {
                    const float* imrow = imc + (size_t)iy * W;
                    const float* wrow  = &s_acc[wv][px][dy * K7];
#pragma unroll
                    for (int dx = 0; dx < K7; ++dx) {
                        int ixx = x + dx - MR;
                        if ((unsigned)ixx < W)
                            acc = fmaf(imrow[ixx], wrow[dx], acc);
                    }
                }
            }
            pred[((size_t)b * 3 + c) * HW + (size_t)y * W + x] = acc;
        }
    }
}

extern "C" void kernel_launch(void* const* d_in, const int* in_sizes, int n_in,
                              void* d_out, int out_size, void* d_ws, size_t ws_size,
                              hipStream_t stream)
{
    const float* im_input = (const float*)d_in[0];   // (16,6,256,256)
    // d_in[1] = im_output: unused by the reference output
    const float* gt       = (const float*)d_in[2];   // (16,2,256,256)
    const float* mker     = (const float*)d_in[3];   // (49,1,7,7)

    float* pred     = (float*)d_out;                 // (16,3,256,256)
    float* mask_out = pred + PRED_ELEMS;             // (16,49,256,256)
    float* axay     = (float*)d_ws;                  // 16 floats / pixel

    axis_mask_kernel<<<(BATCH * HW) / 256, 256, 0, stream>>>(gt, mask_out, axay);

    dim3 grid(W / 16, H / 8, BATCH);
    pred_wmma_kernel<<<grid, 256, 0, stream>>>(im_input, mker, axay, pred);
}